// ROIAlign_27462020890714
// MI455X (gfx1250) — compile-verified
//
#include <hip/hip_runtime.h>
#include <cstdint>

// ROI Align, CDNA5 (gfx1250), wave32.
// x: (4, 256, 128, 128) f32, rois: (K, 5) f32 [b, x1, y1, x2, y2]
// out: (K, 256, 7, 7) f32
//
// Block = 1 ROI, 8 waves; wave w handles channels [w*32, w*32+32).
// Per (roi, channel): stage the <=17x17 bilinear footprint into LDS with
// global_load_async_to_lds_b32 (ASYNCcnt), double-buffered and software-
// pipelined across the channel loop (wait immediate = compile-time batch
// size via template dispatch), then resample 49 cells (16 taps each) from
// LDS. Tap weights/offsets live in registers (channel-invariant).
// No workgroup barriers needed (tiles are wave-private).

#define R_OUT 7
#define R_SR 2
#define R_SCALE 0.0625f
#define R_C 256
#define R_H 128
#define R_W 128
#define TILE_CAP 292   // > 17*17 = 289 max footprint
#define MAX_NIT 10     // ceil(289/32)

__device__ __forceinline__ void corner(float v, int L, int& lo, int& hi,
                                       float& fr, bool& valid) {
    valid = (v > -1.0f) && (v < (float)L);
    float vc = fminf(fmaxf(v, 0.0f), (float)(L - 1));
    float fl = floorf(vc);
    lo = (int)fl;
    hi = min(lo + 1, L - 1);
    fr = vc - fl;
}

// sample coordinate for grid index g (matches reference:
// start + (extent/OUT) * ((g + 0.5)/SR))
__device__ __forceinline__ float sampleCoord(float start, float binsz, int g) {
    return start + binsz * (((float)g + 0.5f) * 0.5f);
}

__device__ __forceinline__ void buildCell(int cell, float syr, float by,
                                          float sxr, float bx, int ylo,
                                          int xlo, int bw, float* w, int* o) {
    int oy = cell / R_OUT;
    int ox = cell - oy * R_OUT;
#pragma unroll
    for (int ssy = 0; ssy < 2; ++ssy) {
        int gy = oy * R_SR + ssy;
        int y0, y1; float fy; bool vy;
        corner(sampleCoord(syr, by, gy), R_H, y0, y1, fy, vy);
#pragma unroll
        for (int ssx = 0; ssx < 2; ++ssx) {
            int gx = ox * R_SR + ssx;
            int x0, x1; float fx; bool vx;
            corner(sampleCoord(sxr, bx, gx), R_W, x0, x1, fx, vx);
            float m = (vy && vx) ? 0.25f : 0.0f;  // pooling mean folded in
            int s = (ssy * 2 + ssx) * 4;
            int ry0 = y0 - ylo, ry1 = y1 - ylo;
            int rx0 = x0 - xlo, rx1 = x1 - xlo;
            o[s + 0] = ry0 * bw + rx0;
            o[s + 1] = ry0 * bw + rx1;
            o[s + 2] = ry1 * bw + rx0;
            o[s + 3] = ry1 * bw + rx1;
            float wy0 = 1.0f - fy, wx0 = 1.0f - fx;
            w[s + 0] = m * wy0 * wx0;
            w[s + 1] = m * wy0 * fx;
            w[s + 2] = m * fy * wx0;
            w[s + 3] = m * fy * fx;
        }
    }
}

// Double-buffered, software-pipelined channel loop. NIT = async-load batch
// size (wave-uniform, ROI-uniform) as a compile-time constant so the
// s_wait_asynccnt immediate can guarantee the older batch has landed.
template <int NIT>
__device__ __forceinline__ void runChannels(
    const float* __restrict__ srcBase, float* __restrict__ out,
    size_t outKBase, int cBase, int lane, int n, int bw,
    uint32_t buf0, uint32_t buf1,
    const float* __restrict__ t0, const float* __restrict__ t1,
    const float* wA, const int* oA, const float* wB, const int* oB,
    int cellA, int cellB) {
    const size_t plane = (size_t)R_H * R_W;

    // channel-invariant per-lane byte offsets for the staging batch
    uint32_t goff[NIT], loff[NIT];
#pragma unroll
    for (int it = 0; it < NIT; ++it) {
        int j = lane + it * 32;
        int jj = min(j, n - 1);  // clamp: keep EXEC full, exact NIT issues
        unsigned row = (unsigned)jj / (unsigned)bw;
        unsigned col = (unsigned)jj - row * (unsigned)bw;
        goff[it] = (row * R_W + col) * 4u;
        loff[it] = (uint32_t)jj * 4u;
    }

    auto issue = [&](int c, uint32_t bufBase) {
        const char* src = (const char*)(srcBase + (size_t)c * plane);
#pragma unroll
        for (int it = 0; it < NIT; ++it) {
            uint32_t laddr = bufBase + loff[it];
            uint64_t gaddr = (uint64_t)(uintptr_t)(src + goff[it]);
            asm volatile("global_load_async_to_lds_b32 %0, %1, off"
                         :
                         : "v"(laddr), "v"(gaddr)
                         : "memory");
        }
    };

    issue(cBase, buf0);  // prologue: channel 0 into buffer 0
    for (int ci = 0; ci < 32; ++ci) {
        if (ci + 1 < 32) {
            // overlap: next channel into the other buffer, then wait only
            // until <= NIT outstanding => batch ci complete (in-order).
            issue(cBase + ci + 1, (ci & 1) ? buf0 : buf1);
            asm volatile("s_wait_asynccnt %0" ::"n"(NIT) : "memory");
        } else {
            asm volatile("s_wait_asynccnt 0" ::: "memory");
        }
        const float* tile = (ci & 1) ? t1 : t0;
        const size_t obase = outKBase + (size_t)(cBase + ci) * (R_OUT * R_OUT);
        float acc = 0.0f;
#pragma unroll
        for (int s = 0; s < 16; ++s) acc = fmaf(wA[s], tile[oA[s]], acc);
        out[obase + cellA] = acc;
        if (cellB < R_OUT * R_OUT) {
            float acc2 = 0.0f;
#pragma unroll
            for (int s = 0; s < 16; ++s) acc2 = fmaf(wB[s], tile[oB[s]], acc2);
            out[obase + cellB] = acc2;
        }
        // ds reads above are consumed by the FMAs (DScnt drained) before the
        // next iteration's volatile asm issues async writes into this buffer
        // two iterations later -> no LDS WAR hazard with the async engine.
    }
}

__global__ __launch_bounds__(256) void roialign_gfx1250_kernel(
    const float* __restrict__ x, const float* __restrict__ rois,
    float* __restrict__ out) {
    __shared__ float tiles[8][2][TILE_CAP];
    const int k = blockIdx.x;
    const int wrp = threadIdx.x >> 5;
    const int lane = threadIdx.x & 31;
    const float* t0 = &tiles[wrp][0][0];
    const float* t1 = &tiles[wrp][1][0];
    const uint32_t buf0 = (uint32_t)(uintptr_t)t0;  // LDS byte offsets
    const uint32_t buf1 = (uint32_t)(uintptr_t)t1;

    // ---- ROI scalars (identical in every lane) ----
    const float r0 = rois[k * 5 + 0];
    const float sxr = rois[k * 5 + 1] * R_SCALE;
    const float syr = rois[k * 5 + 2] * R_SCALE;
    const float exr = rois[k * 5 + 3] * R_SCALE;
    const float eyr = rois[k * 5 + 4] * R_SCALE;
    const int b = (int)r0;
    const float rw = fmaxf(exr - sxr, 1.0f);
    const float rh = fmaxf(eyr - syr, 1.0f);
    const float bx = rw / (float)R_OUT;
    const float by = rh / (float)R_OUT;

    // ---- footprint bbox: sample grid is monotonic, endpoints suffice ----
    int ylo, yhi, xlo, xhi;
    {
        int lo, hi; float fr; bool vd;
        corner(sampleCoord(syr, by, 0), R_H, lo, hi, fr, vd);
        ylo = lo;
        corner(sampleCoord(syr, by, 13), R_H, lo, hi, fr, vd);
        yhi = hi;
        corner(sampleCoord(sxr, bx, 0), R_W, lo, hi, fr, vd);
        xlo = lo;
        corner(sampleCoord(sxr, bx, 13), R_W, lo, hi, fr, vd);
        xhi = hi;
    }
    const int bw = xhi - xlo + 1;  // <= 17
    const int bh = yhi - ylo + 1;  // <= 17
    int n = bh * bw;               // <= 289 <= TILE_CAP
    if (n > TILE_CAP) n = TILE_CAP;  // defensive, cannot trigger
    const int nit = (n + 31) >> 5;   // 1..10, uniform across the block

    // ---- per-lane tap geometry (channel-invariant, kept in VGPRs) ----
    const int cellA = lane;       // 0..31  (< 49 always)
    const int cellB = lane + 32;  // 32..63 (valid if < 49)
    float wA[16]; int oA[16];
    float wB[16]; int oB[16];
    buildCell(cellA, syr, by, sxr, bx, ylo, xlo, bw, wA, oA);
    if (cellB < R_OUT * R_OUT)
        buildCell(cellB, syr, by, sxr, bx, ylo, xlo, bw, wB, oB);

    const size_t plane = (size_t)R_H * R_W;
    const float* srcBase =
        x + (size_t)b * R_C * plane + (size_t)ylo * R_W + xlo;
    const int cBase = wrp * 32;
    const size_t outKBase = (size_t)k * R_C * (R_OUT * R_OUT);

#define RC_CASE(N)                                                          \
    case N:                                                                 \
        runChannels<N>(srcBase, out, outKBase, cBase, lane, n, bw, buf0,    \
                       buf1, t0, t1, wA, oA, wB, oB, cellA, cellB);         \
        break;

    switch (nit) {
        RC_CASE(1)
        RC_CASE(2)
        RC_CASE(3)
        RC_CASE(4)
        RC_CASE(5)
        RC_CASE(6)
        RC_CASE(7)
        RC_CASE(8)
        RC_CASE(9)
        default:
            runChannels<MAX_NIT>(srcBase, out, outKBase, cBase, lane, n, bw,
                                 buf0, buf1, t0, t1, wA, oA, wB, oB, cellA,
                                 cellB);
            break;
    }
#undef RC_CASE
}

extern "C" void kernel_launch(void* const* d_in, const int* in_sizes, int n_in,
                              void* d_out, int out_size, void* d_ws,
                              size_t ws_size, hipStream_t stream) {
    const float* x = (const float*)d_in[0];
    const float* rois = (const float*)d_in[1];
    float* out = (float*)d_out;
    const int K = in_sizes[1] / 5;  // 1000
    roialign_gfx1250_kernel<<<K, 256, 0, stream>>>(x, rois, out);
}